// HouseHolderFlow_44659069944366
// MI455X (gfx1250) — compile-verified
//
#include <hip/hip_runtime.h>

// Householder reflection, B x L row-wise:  out = z - 2*v*(v.z)/(v.v)
// B = 16384, L = 2048, fp32. Pure bandwidth-bound (~402 MB -> ~17 us at 23.3 TB/s).
// Strategy: one 256-thread block per row; async global->LDS staging (CDNA5 path),
// single pass over the data, b128 loads/stores, wave32 shuffle reduction.

typedef float f4 __attribute__((ext_vector_type(4)));

// Pointer types matching __builtin_amdgcn_global_load_async_to_lds_b128:
//   (v4i addrspace(1)* src, v4i addrspace(3)* dst, imm offset, imm cpol)
typedef int vi4 __attribute__((vector_size(16)));
typedef __attribute__((address_space(1))) vi4* g_vi4p;   // global (AS1)
typedef __attribute__((address_space(3))) vi4* l_vi4p;   // LDS (AS3)

#define ROW_LEN 2048
#define THREADS 256
#define NWAVES (THREADS / 32)                       // 8 waves (wave32)
#define CHUNKS_PER_THREAD (ROW_LEN / (THREADS * 4)) // 2 x float4 per array per thread

__global__ __launch_bounds__(THREADS) void
householder_row_kernel(const float* __restrict__ v,
                       const float* __restrict__ z,
                       float* __restrict__ out)
{
    __shared__ float s_v[ROW_LEN];   // 8 KB
    __shared__ float s_z[ROW_LEN];   // 8 KB
    __shared__ float s_red[2 * NWAVES];

    const int    t   = threadIdx.x;
    const size_t row = (size_t)blockIdx.x;

    const float* __restrict__ vrow = v + row * ROW_LEN;
    const float* __restrict__ zrow = z + row * ROW_LEN;
    float*       __restrict__ orow = out + row * ROW_LEN;

    // ---- Phase 1: stage v-row and z-row into LDS ------------------------
#if __has_builtin(__builtin_amdgcn_global_load_async_to_lds_b128)
    // CDNA5 async DMA path: GLOBAL_LOAD_ASYNC_TO_LDS_B128, tracked by ASYNCcnt.
    #pragma unroll
    for (int k = 0; k < CHUNKS_PER_THREAD; ++k) {
        const int e = (t + k * THREADS) * 4;   // float index, 16B-aligned chunk
        __builtin_amdgcn_global_load_async_to_lds_b128(
            (g_vi4p)(vrow + e), (l_vi4p)(&s_v[e]), 0, 0);
        __builtin_amdgcn_global_load_async_to_lds_b128(
            (g_vi4p)(zrow + e), (l_vi4p)(&s_z[e]), 0, 0);
    }
  #if __has_builtin(__builtin_amdgcn_s_wait_asynccnt)
    __builtin_amdgcn_s_wait_asynccnt(0);
  #else
    asm volatile("s_wait_asynccnt 0" ::: "memory");
  #endif
#else
    // Fallback: b128 global load -> b128 LDS store.
    #pragma unroll
    for (int k = 0; k < CHUNKS_PER_THREAD; ++k) {
        const int e = (t + k * THREADS) * 4;
        *(f4*)(&s_v[e]) = *(const f4*)(vrow + e);
        *(f4*)(&s_z[e]) = *(const f4*)(zrow + e);
    }
#endif
    __syncthreads();

    // ---- Phase 2: per-thread partial dot products (read LDS once) -------
    f4 va[CHUNKS_PER_THREAD], za[CHUNKS_PER_THREAD];
    float vz = 0.0f, vv = 0.0f;
    #pragma unroll
    for (int k = 0; k < CHUNKS_PER_THREAD; ++k) {
        const int e = (t + k * THREADS) * 4;
        va[k] = *(const f4*)(&s_v[e]);   // ds_load_b128
        za[k] = *(const f4*)(&s_z[e]);
        vz += va[k].x * za[k].x + va[k].y * za[k].y +
              va[k].z * za[k].z + va[k].w * za[k].w;
        vv += va[k].x * va[k].x + va[k].y * va[k].y +
              va[k].z * va[k].z + va[k].w * va[k].w;
    }

    // ---- Phase 3: wave32 butterfly reduction ----------------------------
    #pragma unroll
    for (int m = 16; m >= 1; m >>= 1) {
        vz += __shfl_xor(vz, m, 32);
        vv += __shfl_xor(vv, m, 32);
    }
    const int wave = t >> 5;
    if ((t & 31) == 0) {
        s_red[wave]          = vz;
        s_red[NWAVES + wave] = vv;
    }
    __syncthreads();

    // Every thread combines the 8 wave partials identically (deterministic
    // order, consistent broadcast value across the block).
    float tvz = 0.0f, tvv = 0.0f;
    #pragma unroll
    for (int w = 0; w < NWAVES; ++w) {
        tvz += s_red[w];
        tvv += s_red[NWAVES + w];
    }
    const float scale = 2.0f * tvz / tvv;

    // ---- Phase 4: out = z - scale * v, b128 stores ----------------------
    #pragma unroll
    for (int k = 0; k < CHUNKS_PER_THREAD; ++k) {
        const int e = (t + k * THREADS) * 4;
        f4 o = za[k] - scale * va[k];
        *(f4*)(orow + e) = o;               // global_store_b128
    }
}

extern "C" void kernel_launch(void* const* d_in, const int* in_sizes, int n_in,
                              void* d_out, int out_size, void* d_ws, size_t ws_size,
                              hipStream_t stream) {
    (void)n_in; (void)out_size; (void)d_ws; (void)ws_size;
    const float* v = (const float*)d_in[0];
    const float* z = (const float*)d_in[1];
    float* out     = (float*)d_out;

    const int B = in_sizes[0] / ROW_LEN;   // 16384 rows
    householder_row_kernel<<<dim3(B), dim3(THREADS), 0, stream>>>(v, z, out);
}